// rSigKer_2843268350159
// MI455X (gfx1250) — compile-verified
//
#include <hip/hip_runtime.h>
#include <stdint.h>

// MI455X / gfx1250, wave32. Persistent-grid random-signature CDE scan + Gram.
// Per step: z_new = z + P·G^T as a K=16*516 (padded) f32 WMMA GEMM.
// 8 row-teams (16 batch rows each) x 16 column slices = 128 WGs, 4 waves/WG.

typedef __attribute__((ext_vector_type(2))) float v2f;
typedef __attribute__((ext_vector_type(8))) float v8f;

#define DCH 16
#define NH 512
#define TT 128
#define NP1 513
#define ROWS 16
#define SLICES 16
#define COLS_PER_WG 32
#define TEAMS 8
#define NBLOCKS (TEAMS * SLICES)
#define NTHREADS 128
#define ZSTRIDE 513        // global z row stride (floats)
#define LDS_ZSTRIDE 516    // padded LDS za row stride (multiple of 4)

__device__ __forceinline__ v8f wmma_f32(v2f a, v2f b, v8f c) {
  return __builtin_amdgcn_wmma_f32_16x16x4_f32(false, a, false, b, (short)0, c,
                                               false, false);
}

// Sense-reversing barrier over `members` workgroups (agent-scope atomics).
__device__ __forceinline__ void gbar(uint32_t* cnt, uint32_t* phase,
                                     uint32_t members, uint32_t myphase) {
  __threadfence();  // release: flush our z writes to device scope
  __syncthreads();
  if (threadIdx.x == 0) {
    uint32_t old = __hip_atomic_fetch_add(cnt, 1u, __ATOMIC_ACQ_REL,
                                          __HIP_MEMORY_SCOPE_AGENT);
    if (old == members - 1u) {
      __hip_atomic_store(cnt, 0u, __ATOMIC_RELAXED, __HIP_MEMORY_SCOPE_AGENT);
      __hip_atomic_fetch_add(phase, 1u, __ATOMIC_RELEASE,
                             __HIP_MEMORY_SCOPE_AGENT);
    } else {
      while (__hip_atomic_load(phase, __ATOMIC_ACQUIRE,
                               __HIP_MEMORY_SCOPE_AGENT) <= myphase) {
        __builtin_amdgcn_s_sleep(2);
      }
    }
  }
  __syncthreads();
  __threadfence();  // acquire: invalidate L0 so fresh z is visible
}

__global__ __launch_bounds__(NTHREADS) void rsig_scan_gram(
    const float* __restrict__ X, const float* __restrict__ Y,
    const float* __restrict__ Gm, const float* __restrict__ z0,
    float* __restrict__ out, float* __restrict__ zbuf,
    uint32_t* __restrict__ bar) {
  __shared__ float za_lds[ROWS * LDS_ZSTRIDE];  // 33024 B
  __shared__ float dx_lds[ROWS * DCH];          // 1024 B
  __shared__ float red_lds[2 * 32 * 8];         // 2048 B

  const int tid = threadIdx.x;
  const int wid = tid >> 5;
  const int lane = tid & 31;
  const int hi = lane >> 4;    // half-wave select
  const int l15 = lane & 15;
  const int aoff = 2 * hi;     // K offset owned by this half-wave
  const int blk = blockIdx.x;
  const int team = blk >> 4;   // 0..7 (0-3: X rows, 4-7: Y rows)
  const int slice = blk & 15;  // 0..15 column slice
  const int rowbase = team * ROWS;  // row base in 128-row z buffer
  const float* src = (team < 4) ? X : Y;
  const int rbase_s = (team < 4 ? team : team - 4) * ROWS;

  uint32_t* tcnt = bar + team;
  uint32_t* tphase = bar + 8 + team;
  uint32_t* gcnt = bar + 16;
  uint32_t* gphase = bar + 17;
  uint32_t tph = 0;

  float* zb0 = zbuf;
  float* zb1 = zbuf + 128 * ZSTRIDE;

  // ---- init: broadcast z0 into buffer 0; ones-column in both buffers ----
  for (int idx = tid; idx < ROWS * COLS_PER_WG; idx += NTHREADS) {
    int r = idx >> 5, c = idx & 31;
    int n = slice * COLS_PER_WG + c;
    zb0[(rowbase + r) * ZSTRIDE + n] = z0[n];
  }
  if (slice == SLICES - 1) {
    for (int r = tid; r < ROWS; r += NTHREADS) {
      zb0[(rowbase + r) * ZSTRIDE + NH] = 1.0f;
      zb1[(rowbase + r) * ZSTRIDE + NH] = 1.0f;
    }
  }
  gbar(tcnt, tphase, SLICES, tph);
  tph++;

  const int tile = wid & 1;  // which 16-col tile of this WG
  const int kh = wid >> 1;   // which half of the d-range
  const int n_tile0 = slice * COLS_PER_WG + tile * 16;
  const int n_b = n_tile0 + l15;  // B-matrix column for this lane
  const int arow = l15;           // A-matrix row for this lane

  float* cur = zb0;
  float* nxt = zb1;

  for (int t = 0; t < TT - 1; ++t) {
    // stage za tile (16 x 516, cols >512 zero) into LDS
    {
      int r = tid >> 3, tl = tid & 7;
      const float* zr = cur + (rowbase + r) * ZSTRIDE;
      float* dst = za_lds + r * LDS_ZSTRIDE;
      for (int m = tl; m < LDS_ZSTRIDE; m += 8)
        dst[m] = (m < NP1) ? zr[m] : 0.0f;
    }
    // stage dx tile (16 x 16)
    for (int idx = tid; idx < ROWS * DCH; idx += NTHREADS) {
      int r = idx >> 4, d = idx & 15;
      const float* xp = src + ((size_t)(rbase_s + r) * TT + t) * DCH + d;
      dx_lds[idx] = xp[DCH] - xp[0];
    }
    __syncthreads();

    v8f acc0 = {};
    v8f acc1 = {};
    const float* zrow = za_lds + arow * LDS_ZSTRIDE;
    for (int d = kh * 8; d < kh * 8 + 8; ++d) {
      float xd = dx_lds[arow * DCH + d];
      const float* grow = Gm + (size_t)(d * NH + n_b) * NP1;
      for (int m0 = 0; m0 < 512; m0 += 8) {
        v2f zp0 = *(const v2f*)(zrow + m0 + aoff);
        v2f zp1 = *(const v2f*)(zrow + m0 + 4 + aoff);
        v2f a0 = zp0 * xd;
        v2f a1 = zp1 * xd;
        v2f b0 = {grow[m0 + aoff], grow[m0 + aoff + 1]};
        v2f b1 = {grow[m0 + 4 + aoff], grow[m0 + 4 + aoff + 1]};
        acc0 = wmma_f32(a0, b0, acc0);
        acc1 = wmma_f32(a1, b1, acc1);
      }
      {  // tail chunk: k = 512..515; za pads are 0, clamp Gamma index to 512
        v2f zt = *(const v2f*)(zrow + 512 + aoff);
        v2f a0 = zt * xd;
        int mi0 = 512 + aoff;
        v2f b0 = {grow[mi0 > 512 ? 512 : mi0], grow[512]};
        acc0 = wmma_f32(a0, b0, acc0);
      }
    }
    v8f acc = acc0 + acc1;

    // cross-wave K reduction through LDS, then z update
    __syncthreads();
    if (kh == 1) {
#pragma unroll
      for (int j = 0; j < 8; ++j)
        red_lds[(tile * 32 + lane) * 8 + j] = acc[j];
    }
    __syncthreads();
    if (kh == 0) {
#pragma unroll
      for (int j = 0; j < 8; ++j)
        acc[j] += red_lds[(tile * 32 + lane) * 8 + j];
#pragma unroll
      for (int j = 0; j < 8; ++j) {
        int r = j + 8 * hi;  // D-matrix row layout: VGPR j -> M=j / M=j+8
        int n = n_tile0 + l15;
        float zold = za_lds[r * LDS_ZSTRIDE + n];
        nxt[(rowbase + r) * ZSTRIDE + n] = zold + acc[j];
      }
    }
    gbar(tcnt, tphase, SLICES, tph);
    tph++;
    float* tmp = cur;
    cur = nxt;
    nxt = tmp;
  }

  // ---- grid-wide barrier, then Gram: G = zX * zY^T / 512 ----
  gbar(gcnt, gphase, NBLOCKS, 0);
  const float* zf = cur;  // final z_T for all 128 rows
  if (blk < 16 && wid == 0) {
    int gi = blk >> 2, gj = blk & 3;
    const float* ar = zf + (size_t)(gi * 16 + l15) * ZSTRIDE;
    const float* br = zf + (size_t)(64 + gj * 16 + l15) * ZSTRIDE;
    v8f c0 = {};
    v8f c1 = {};
    for (int k0 = 0; k0 < NH; k0 += 8) {
      v2f a0 = {ar[k0 + aoff], ar[k0 + aoff + 1]};
      v2f b0 = {br[k0 + aoff], br[k0 + aoff + 1]};
      v2f a1 = {ar[k0 + 4 + aoff], ar[k0 + 4 + aoff + 1]};
      v2f b1 = {br[k0 + 4 + aoff], br[k0 + 4 + aoff + 1]};
      c0 = wmma_f32(a0, b0, c0);
      c1 = wmma_f32(a1, b1, c1);
    }
    v8f c = c0 + c1;
#pragma unroll
    for (int j = 0; j < 8; ++j) {
      int i = gi * 16 + j + 8 * hi;
      int jj = gj * 16 + l15;
      out[i * 64 + jj] = c[j] * (1.0f / 512.0f);
    }
  }
}

extern "C" void kernel_launch(void* const* d_in, const int* in_sizes, int n_in,
                              void* d_out, int out_size, void* d_ws,
                              size_t ws_size, hipStream_t stream) {
  const float* X = (const float*)d_in[0];   // (64,128,16)
  const float* Y = (const float*)d_in[1];   // (64,128,16)
  const float* Gm = (const float*)d_in[2];  // (16,512,513)
  const float* z0 = (const float*)d_in[3];  // (512)
  float* outp = (float*)d_out;              // (64,64)

  // workspace: z double-buffer (2*128*513 f32 = 525312 B) + barrier words
  float* zbuf = (float*)d_ws;
  uint32_t* bar = (uint32_t*)((char*)d_ws + (size_t)2 * 128 * ZSTRIDE * 4);
  hipMemsetAsync(bar, 0, 32 * sizeof(uint32_t), stream);

  rsig_scan_gram<<<NBLOCKS, NTHREADS, 0, stream>>>(X, Y, Gm, z0, outp, zbuf,
                                                   bar);
}